// GCN_26044681683459
// MI455X (gfx1250) — compile-verified
//
#include <hip/hip_runtime.h>

typedef __attribute__((ext_vector_type(2))) float v2f;
typedef __attribute__((ext_vector_type(8))) float v8f;

#define IN_C 64
#define POSD 16
#define KDIM1 (IN_C + POSD)   // 80
#define HID 64
#define OUTD 32

// ---------------- small utility kernels ----------------

__global__ void k_zero(float* __restrict__ p, int n) {
  int i = blockIdx.x * blockDim.x + threadIdx.x;
  if (i < n) p[i] = 0.0f;
}

// degree count over dst edges (self-loop +1 added in k_dinv)
__global__ void k_deg(const int* __restrict__ dst, float* __restrict__ deg, int E) {
  int i = blockIdx.x * blockDim.x + threadIdx.x;
  if (i < E) atomicAdd(&deg[dst[i]], 1.0f);
}

// dinv = rsqrt(deg + 1)   (deg >= 1 always due to self loop)
__global__ void k_dinv(float* __restrict__ d, int n) {
  int i = blockIdx.x * blockDim.x + threadIdx.x;
  if (i < n) d[i] = rsqrtf(d[i] + 1.0f);
}

// ---------------- WMMA GEMM 1: hw1 = concat(x,pos) @ W1 ----------------
// one wave per 16-row tile, 4 accumulators cover all 64 output cols.
// Epilogue also writes the self-loop-initialized aggregation buffer:
//   h1agg[r][c] = hw1[r][c] * dinv[r]^2
__global__ __launch_bounds__(256) void k_gemm1(
    const float* __restrict__ x, const float* __restrict__ pos,
    const float* __restrict__ W1, const float* __restrict__ dinv,
    float* __restrict__ hw1, float* __restrict__ h1agg, int N) {
  const int lane = threadIdx.x & 31;
  const int wave = threadIdx.x >> 5;
  const int rowBase = (blockIdx.x * 8 + wave) * 16;
  if (rowBase >= N) return;                 // wave-uniform
  const int kh  = lane >> 4;                // 0 or 1 (K half)
  const int n16 = lane & 15;
  int row = rowBase + n16;
  if (row >= N) row = N - 1;                // clamp reads (stores are guarded)

  v8f acc0 = {}, acc1 = {}, acc2 = {}, acc3 = {};
  for (int k = 0; k < KDIM1; k += 4) {
    const int ka = k + 2 * kh;
    v2f a;
    a.x = (ka     < IN_C) ? x[(long)row * IN_C + ka]       : pos[(long)row * POSD + (ka     - IN_C)];
    a.y = (ka + 1 < IN_C) ? x[(long)row * IN_C + ka + 1]   : pos[(long)row * POSD + (ka + 1 - IN_C)];
    const float* wb = W1 + (long)ka * HID + n16;
    v2f b0; b0.x = wb[0];  b0.y = wb[HID];
    v2f b1; b1.x = wb[16]; b1.y = wb[HID + 16];
    v2f b2; b2.x = wb[32]; b2.y = wb[HID + 32];
    v2f b3; b3.x = wb[48]; b3.y = wb[HID + 48];
    acc0 = __builtin_amdgcn_wmma_f32_16x16x4_f32(false, a, false, b0, (short)0, acc0, false, false);
    acc1 = __builtin_amdgcn_wmma_f32_16x16x4_f32(false, a, false, b1, (short)0, acc1, false, false);
    acc2 = __builtin_amdgcn_wmma_f32_16x16x4_f32(false, a, false, b2, (short)0, acc2, false, false);
    acc3 = __builtin_amdgcn_wmma_f32_16x16x4_f32(false, a, false, b3, (short)0, acc3, false, false);
  }
#pragma unroll
  for (int i = 0; i < 8; ++i) {
    const int r = rowBase + i + 8 * kh;     // C/D layout: vgpr i -> M=i (lanes<16), M=i+8 (lanes>=16)
    if (r < N) {
      const float dv = dinv[r];
      const float sw = dv * dv;
      const long base = (long)r * HID + n16;
      hw1[base +  0] = acc0[i];  h1agg[base +  0] = acc0[i] * sw;
      hw1[base + 16] = acc1[i];  h1agg[base + 16] = acc1[i] * sw;
      hw1[base + 32] = acc2[i];  h1agg[base + 32] = acc2[i] * sw;
      hw1[base + 48] = acc3[i];  h1agg[base + 48] = acc3[i] * sw;
    }
  }
}

// ---------------- WMMA GEMM 2: hw2 = h1 @ W2 ----------------
__global__ __launch_bounds__(256) void k_gemm2(
    const float* __restrict__ h, const float* __restrict__ W2,
    const float* __restrict__ dinv, float* __restrict__ hw2,
    float* __restrict__ zinit, int N) {
  const int lane = threadIdx.x & 31;
  const int wave = threadIdx.x >> 5;
  const int rowBase = (blockIdx.x * 8 + wave) * 16;
  if (rowBase >= N) return;
  const int kh  = lane >> 4;
  const int n16 = lane & 15;
  int row = rowBase + n16;
  if (row >= N) row = N - 1;

  v8f acc0 = {}, acc1 = {};
  for (int k = 0; k < HID; k += 4) {
    const int ka = k + 2 * kh;
    v2f a;
    a.x = h[(long)row * HID + ka];
    a.y = h[(long)row * HID + ka + 1];
    const float* wb = W2 + (long)ka * OUTD + n16;
    v2f b0; b0.x = wb[0];  b0.y = wb[OUTD];
    v2f b1; b1.x = wb[16]; b1.y = wb[OUTD + 16];
    acc0 = __builtin_amdgcn_wmma_f32_16x16x4_f32(false, a, false, b0, (short)0, acc0, false, false);
    acc1 = __builtin_amdgcn_wmma_f32_16x16x4_f32(false, a, false, b1, (short)0, acc1, false, false);
  }
#pragma unroll
  for (int i = 0; i < 8; ++i) {
    const int r = rowBase + i + 8 * kh;
    if (r < N) {
      const float dv = dinv[r];
      const float sw = dv * dv;
      const long base = (long)r * OUTD + n16;
      hw2[base +  0] = acc0[i];  zinit[base +  0] = acc0[i] * sw;
      hw2[base + 16] = acc1[i];  zinit[base + 16] = acc1[i] * sw;
    }
  }
}

// ---------------- edge scatter-add: agg[dst] += feat[src] * dinv[s]*dinv[d] ----
// one thread per (edge, channel); C = 1 << shift channels
__global__ void k_scatter(const int* __restrict__ src, const int* __restrict__ dst,
                          const float* __restrict__ dinv, const float* __restrict__ feat,
                          float* __restrict__ agg, long total, int shift) {
  long t = (long)blockIdx.x * blockDim.x + threadIdx.x;
  if (t >= total) return;
  const int e = (int)(t >> shift);
  const int c = (int)(t & ((1L << shift) - 1));
  const int s = src[e];
  const int d = dst[e];
  const float w = dinv[s] * dinv[d];
  atomicAdd(&agg[((long)d << shift) + c], feat[((long)s << shift) + c] * w);
}

// ---------------- bias (+ optional relu), in place ----------------
__global__ void k_bias_act(float* __restrict__ h, const float* __restrict__ b,
                           long total, int mask, int relu) {
  long i = (long)blockIdx.x * blockDim.x + threadIdx.x;
  if (i >= total) return;
  float v = h[i] + b[(int)(i & mask)];
  if (relu) v = fmaxf(v, 0.0f);
  h[i] = v;
}

// ---------------- link predictor: pred[e] = [z[s],z[d]] . Wl + bl ----------------
__global__ void k_link(const float* __restrict__ z, const int* __restrict__ eli,
                       const float* __restrict__ Wl, const float* __restrict__ bl,
                       float* __restrict__ pred, int EL) {
  int e = blockIdx.x * blockDim.x + threadIdx.x;
  if (e >= EL) return;
  const int s = eli[e];
  const int d = eli[EL + e];
  const float* zs = z + (long)s * OUTD;
  const float* zd = z + (long)d * OUTD;
  float acc = bl[0];
#pragma unroll
  for (int c = 0; c < OUTD; ++c)
    acc = fmaf(zs[c], Wl[c], fmaf(zd[c], Wl[OUTD + c], acc));
  pred[e] = acc;
}

// ---------------- launch ----------------
extern "C" void kernel_launch(void* const* d_in, const int* in_sizes, int n_in,
                              void* d_out, int out_size, void* d_ws, size_t ws_size,
                              hipStream_t stream) {
  const float* x   = (const float*)d_in[0];
  const float* pos = (const float*)d_in[1];
  const float* W1  = (const float*)d_in[2];
  const float* b1  = (const float*)d_in[3];
  const float* W2  = (const float*)d_in[4];
  const float* b2  = (const float*)d_in[5];
  const float* Wl  = (const float*)d_in[6];
  const float* bl  = (const float*)d_in[7];
  const int*   ei  = (const int*)d_in[8];
  const int*   eli = (const int*)d_in[9];

  const int N  = in_sizes[0] / IN_C;
  const int E  = in_sizes[8] / 2;
  const int EL = in_sizes[9] / 2;

  float* z    = (float*)d_out;             // [N, OUTD]
  float* pred = z + (long)N * OUTD;        // [EL]

  float* dinv  = (float*)d_ws;             // [N]
  float* hw1   = dinv + N;                 // [N, HID]
  float* h1    = hw1 + (long)N * HID;      // [N, HID] (agg -> relu in place)
  float* hw2   = h1  + (long)N * HID;      // [N, OUTD]

  const int* esrc = ei;
  const int* edst = ei + E;

  const int B = 256;

  // degree / normalization
  k_zero<<<(N + B - 1) / B, B, 0, stream>>>(dinv, N);
  k_deg <<<(E + B - 1) / B, B, 0, stream>>>(edst, dinv, E);
  k_dinv<<<(N + B - 1) / B, B, 0, stream>>>(dinv, N);

  // layer 1
  const int rowTiles = (N + 15) / 16;
  const int gblocks  = (rowTiles + 7) / 8;
  k_gemm1<<<gblocks, B, 0, stream>>>(x, pos, W1, dinv, hw1, h1, N);
  {
    long tot = (long)E * HID;
    k_scatter<<<(int)((tot + B - 1) / B), B, 0, stream>>>(esrc, edst, dinv, hw1, h1, tot, 6);
    long nh = (long)N * HID;
    k_bias_act<<<(int)((nh + B - 1) / B), B, 0, stream>>>(h1, b1, nh, HID - 1, 1);
  }

  // layer 2 (accumulate straight into z region of d_out)
  k_gemm2<<<gblocks, B, 0, stream>>>(h1, W2, dinv, hw2, z, N);
  {
    long tot = (long)E * OUTD;
    k_scatter<<<(int)((tot + B - 1) / B), B, 0, stream>>>(esrc, edst, dinv, hw2, z, tot, 5);
    long nz = (long)N * OUTD;
    k_bias_act<<<(int)((nz + B - 1) / B), B, 0, stream>>>(z, b2, nz, OUTD - 1, 0);
  }

  // link prediction
  k_link<<<(EL + B - 1) / B, B, 0, stream>>>(z, eli, Wl, bl, pred, EL);
}